// hashedLayer_39487929319938
// MI455X (gfx1250) — compile-verified
//
#include <hip/hip_runtime.h>

typedef __attribute__((ext_vector_type(2))) float v2f;
typedef __attribute__((ext_vector_type(8))) float v8f;

#define BATCH   262144
#define FAN_IN  128
#define FAN_OUT 128
#define HJ      129              // FAN_IN + 1 (bias column appended)
#define NTILES  (BATCH / 16)     // 16384 batch tiles of 16 rows

// out[b, i] = sum_{j<128} a[b,j] * W[hash_idx[i,j]]  +  W[hash_idx[i,128]]
// GEMM M=BATCH, N=128, K=128 (+bias), fp32, via V_WMMA_F32_16X16X4_F32.
__global__ __launch_bounds__(256) void hashed_layer_wmma(
    const float* __restrict__ a,
    const float* __restrict__ W,
    const int*   __restrict__ hash_idx,   // int32 per harness convention
    float* __restrict__ out)
{
    const int tid   = threadIdx.x;
    const int wave  = tid >> 5;          // 0..7 : which 16-column N-tile
    const int lane  = tid & 31;
    const int lrow  = lane & 15;         // M index (A) / N index (B, C/D)
    const int khalf = lane >> 4;         // 0/1 : K half selector

    const int n = wave * 16 + lrow;      // output column this lane owns in B/C

    // ---- Gather B fragments (Wmat^T) into registers, once per wave ----
    // B (4x16 f32) layout: element v of lane L holds B[K = v + 2*(L>>4)][N = L&15]
    // B[k][n] = Wmat[n][k] = W[hash_idx[n*129 + k]]
    const int* __restrict__ hrow = hash_idx + n * HJ;
    v2f bfrag[32];
#pragma unroll
    for (int k = 0; k < 32; ++k) {
        const int j = 4 * k + 2 * khalf;
        v2f b;
        b.x = W[hrow[j]];
        b.y = W[hrow[j + 1]];
        bfrag[k] = b;
    }
    // Bias (ones column j = 128); depends only on N -> per-lane constant.
    const float bias = W[hrow[FAN_IN]];

    // ---- Grid-stride loop over 16-row batch tiles ----
    for (int t = blockIdx.x; t < NTILES; t += gridDim.x) {
        const int row_base = t * 16;

        // A (16x4 f32) layout: element v of lane L holds
        //   A[M = L&15][K = v + 2*(L>>4)]  -> contiguous float pair per lane.
        const float* __restrict__ arow =
            a + (size_t)(row_base + lrow) * FAN_IN + 2 * khalf;

        // Prefetch next tile's A row into cache.
        const size_t next_row = (size_t)(row_base + (int)gridDim.x * 16 + lrow);
        if (next_row < (size_t)BATCH)
            __builtin_prefetch(a + next_row * FAN_IN, 0, 1);

        v8f c = {bias, bias, bias, bias, bias, bias, bias, bias};
#pragma unroll
        for (int k = 0; k < 32; ++k) {
            const v2f av = *(const v2f*)(arow + 4 * k);   // global_load_b64
            // 8 args: (neg_a, A, neg_b, B, c_mod, C, reuse_a, reuse_b)
            c = __builtin_amdgcn_wmma_f32_16x16x4_f32(
                    false, av, false, bfrag[k], (short)0, c, false, false);
        }

        // C/D (16x16 f32) layout: element v of lane L -> M = v + 8*(L>>4), N = L&15
        float* __restrict__ orow =
            out + (size_t)(row_base + 8 * khalf) * FAN_OUT + n;
#pragma unroll
        for (int v = 0; v < 8; ++v)
            orow[(size_t)v * FAN_OUT] = c[v];
    }
}

extern "C" void kernel_launch(void* const* d_in, const int* in_sizes, int n_in,
                              void* d_out, int out_size, void* d_ws, size_t ws_size,
                              hipStream_t stream) {
    (void)in_sizes; (void)n_in; (void)d_ws; (void)ws_size; (void)out_size;
    const float* a        = (const float*)d_in[0];
    const float* W        = (const float*)d_in[1];
    const int*   hash_idx = (const int*)d_in[2];
    float*       out      = (float*)d_out;

    // 16384 batch tiles; 2048 blocks x 8 waves keeps every WGP busy while each
    // block's 8 waves share the same A tiles through WGP$/L2.
    dim3 grid(2048), block(256);
    hipLaunchKernelGGL(hashed_layer_wmma, grid, block, 0, stream,
                       a, W, hash_idx, out);
}